// TypeLoss_84610855731460
// MI455X (gfx1250) — compile-verified
//
#include <hip/hip_runtime.h>
#include <stdint.h>

typedef __attribute__((ext_vector_type(2))) float v2f;
typedef __attribute__((ext_vector_type(8))) float v8f;

// pred -> relation type class: 0 none, 1 support, 2 proximity, 3 comparative
__constant__ int c_table[64] = {0,1,2,2,2,2,2,2,3,3,3,3,3,3,
                                1,1,1,1,1,1,1,1,1,1,1,1,1}; // 27..63 zero-filled

// Sum of 32 per-lane floats, result broadcast to all lanes.
// Uses V_WMMA_F32_16X16X4_F32 with an all-ones A matrix: D[i][j] = colsum_j(B),
// which is layout-permutation-invariant for a full sum. EXEC must be all 1s
// (call only from converged code).
__device__ __forceinline__ float wave_sum32(float v) {
#if __has_builtin(__builtin_amdgcn_wmma_f32_16x16x4_f32)
    v2f a; a[0] = 1.0f; a[1] = 1.0f;          // 16x4 ones
    v2f b; b[0] = v;    b[1] = 0.0f;          // 4x16 holds the 32 lane values
    v8f c = {};
    c = __builtin_amdgcn_wmma_f32_16x16x4_f32(false, a, false, b,
                                              (short)0, c, false, false);
    float s = c[0];                            // row of column-sums (dup in both halves)
#else
    float s = v + __shfl_xor(v, 16, 32);
#endif
    s += __shfl_xor(s, 8, 32);
    s += __shfl_xor(s, 4, 32);
    s += __shfl_xor(s, 2, 32);
    s += __shfl_xor(s, 1, 32);
    return s;
}

__global__ void zero_mask_kernel(uint32_t* __restrict__ mask, int n_words) {
    int i = blockIdx.x * blockDim.x + threadIdx.x;
    int stride = gridDim.x * blockDim.x;
    for (; i < n_words; i += stride) mask[i] = 0u;
}

__global__ void scatter_kernel(const int* __restrict__ rel_gt,
                               uint32_t* __restrict__ mask,
                               int n_rel, int insnum) {
    int r = blockIdx.x * blockDim.x + threadIdx.x;
    if (r >= n_rel) return;
    int i = rel_gt[3 * r + 0];
    int j = rel_gt[3 * r + 1];
    int p = rel_gt[3 * r + 2];
    if (i == j) return;                        // mode='drop' case
    p = min(max(p, 0), 63);
    int t = c_table[p];
    int flat = i * (insnum - 1) + j - ((j > i) ? 1 : 0);
    atomicOr(&mask[flat >> 3], 1u << (((flat & 7) << 2) + t));
}

__global__ void loss_kernel(const float4* __restrict__ logits,
                            const uint32_t* __restrict__ mask,
                            const float* __restrict__ alpha,
                            float* __restrict__ partials, int n_pairs) {
    const float a0 = alpha[0], a1 = alpha[1], a2 = alpha[2], a3 = alpha[3];
    float acc = 0.0f;
    const int stride = gridDim.x * blockDim.x;
    for (int i = blockIdx.x * blockDim.x + threadIdx.x; i < n_pairs; i += stride) {
        __builtin_prefetch(logits + i + stride, 0, 0);   // global_prefetch_b8
        float4 x = logits[i];                            // one row == 16B == b128
        uint32_t nib = (mask[i >> 3] >> ((i & 7) << 2)) & 0xFu;
        if (nib == 0u) nib = 1u;                         // empty row -> class 0
        float m  = fmaxf(fmaxf(x.x, x.y), fmaxf(x.z, x.w));
        float e0 = __expf(x.x - m), e1 = __expf(x.y - m),
              e2 = __expf(x.z - m), e3 = __expf(x.w - m);
        float s  = e0 + e1 + e2 + e3;
        float num = 0.0f, ar = 0.0f;
        if (nib & 1u) { num += e0; ar += a0; }
        if (nib & 2u) { num += e1; ar += a1; }
        if (nib & 4u) { num += e2; ar += a2; }
        if (nib & 8u) { num += e3; ar += a3; }
        float p  = num / s;
        float om = 1.0f - p;
        acc += -ar * om * om * __logf(p);
    }
    __shared__ float sdata[8];
    float ws = wave_sum32(acc);                          // WMMA-based wave reduce
    int lane = threadIdx.x & 31, wave = threadIdx.x >> 5;
    if (lane == 0) sdata[wave] = ws;
    __syncthreads();
    if (threadIdx.x == 0) {
        float t = 0.0f;
        int nw = blockDim.x >> 5;
        for (int w = 0; w < nw; ++w) t += sdata[w];
        partials[blockIdx.x] = t;
    }
}

__global__ void final_kernel(const float* __restrict__ partials, int n_partials,
                             float* __restrict__ out, float inv_n) {
    __shared__ float sdata[8];
    int lane = threadIdx.x & 31, wave = threadIdx.x >> 5;
#if __has_builtin(__builtin_amdgcn_wmma_f32_16x16x4_f32)
    v2f a; a[0] = 1.0f; a[1] = 1.0f;
    v8f c = {};
    int wstride = (blockDim.x >> 5) * 64;
    for (int base = wave * 64; base < n_partials; base += wstride) {
        int i0 = base + lane, i1 = base + 32 + lane;
        v2f b;
        b[0] = (i0 < n_partials) ? partials[i0] : 0.0f;
        b[1] = (i1 < n_partials) ? partials[i1] : 0.0f;
        c = __builtin_amdgcn_wmma_f32_16x16x4_f32(false, a, false, b,
                                                  (short)0, c, false, false);
    }
    float s = c[0];
#else
    float s = 0.0f;
    for (int i = threadIdx.x; i < n_partials; i += blockDim.x) s += partials[i];
    s += __shfl_xor(s, 16, 32);
#endif
    s += __shfl_xor(s, 8, 32);
    s += __shfl_xor(s, 4, 32);
    s += __shfl_xor(s, 2, 32);
    s += __shfl_xor(s, 1, 32);
    if (lane == 0) sdata[wave] = s;
    __syncthreads();
    if (threadIdx.x == 0) {
        float t = 0.0f;
        int nw = blockDim.x >> 5;
        for (int w = 0; w < nw; ++w) t += sdata[w];
        out[0] = t * inv_n;
    }
}

extern "C" void kernel_launch(void* const* d_in, const int* in_sizes, int n_in,
                              void* d_out, int out_size, void* d_ws, size_t ws_size,
                              hipStream_t stream) {
    (void)n_in; (void)out_size; (void)ws_size;
    const float4* type_output = (const float4*)d_in[0];
    const int*    rel_gt      = (const int*)d_in[2];
    const float*  pred_w      = (const float*)d_in[3];

    const int insnum  = in_sizes[1];        // obj_gt length
    const int n_rel   = in_sizes[2] / 3;    // rel_gt rows
    const int n_pairs = in_sizes[0] / 4;    // type_output rows == insnum*(insnum-1)

    // Workspace layout: [ packed 4-bit masks (8 rows per uint32) | block partials ]
    uint32_t* mask = (uint32_t*)d_ws;
    const int n_words = (n_pairs + 7) / 8;
    size_t mask_bytes = (((size_t)n_words * 4) + 255) & ~(size_t)255;
    float* partials = (float*)((char*)d_ws + mask_bytes);

    const int TPB = 256;                    // 8 wave32 per block
    const int LOSS_BLOCKS = 2048;

    zero_mask_kernel<<<512, TPB, 0, stream>>>(mask, n_words);
    scatter_kernel<<<(n_rel + TPB - 1) / TPB, TPB, 0, stream>>>(rel_gt, mask,
                                                                n_rel, insnum);
    loss_kernel<<<LOSS_BLOCKS, TPB, 0, stream>>>(type_output, mask, pred_w,
                                                 partials, n_pairs);
    final_kernel<<<1, TPB, 0, stream>>>(partials, LOSS_BLOCKS, (float*)d_out,
                                        1.0f / (float)n_pairs);
}